// SingleMambaBlock_1915555414509
// MI455X (gfx1250) — compile-verified
//
#include <hip/hip_runtime.h>
#include <math.h>

#define BB 4
#define LL 4096
#define DM 512
#define EE 512
#define NS 8
#define RR 32
#define ROWS (BB*LL)          /* 16384 */
#define LDXZ 1024
#define LDBC 48
#define CHUNK 64
#define NCH (LL/CHUNK)        /* 64 */

typedef __attribute__((ext_vector_type(2))) float v2f;
typedef __attribute__((ext_vector_type(8))) float v8f;

__device__ __forceinline__ float waveReduceSum(float v) {
#pragma unroll
  for (int off = 16; off > 0; off >>= 1) v += __shfl_xor(v, off, 32);
  return v;
}

__device__ __forceinline__ float silu_f(float v) {
  return v * (1.f / (1.f + __expf(-v)));
}

// ---------------- LayerNorm: h = LN(x)*w + b ----------------
__global__ void ln_kernel(const float* __restrict__ x, const float* __restrict__ w,
                          const float* __restrict__ b, float* __restrict__ h) {
  int row = blockIdx.x;
  const float* xr = x + (size_t)row * DM;
  int t = threadIdx.x;                    // 256 threads, 2 elems each
  float a0 = xr[t], a1 = xr[t + 256];
  float s  = a0 + a1;
  float ss = a0*a0 + a1*a1;
  __shared__ float ls[8], lss[8];
  s = waveReduceSum(s); ss = waveReduceSum(ss);
  int wid = t >> 5, lane = t & 31;
  if (lane == 0) { ls[wid] = s; lss[wid] = ss; }
  __syncthreads();
  if (wid == 0) {
    float s2  = (lane < 8) ? ls[lane]  : 0.f;
    float ss2 = (lane < 8) ? lss[lane] : 0.f;
    s2 = waveReduceSum(s2); ss2 = waveReduceSum(ss2);
    if (lane == 0) { ls[0] = s2; lss[0] = ss2; }
  }
  __syncthreads();
  float mean = ls[0] * (1.f/DM);
  float var  = lss[0] * (1.f/DM) - mean*mean;
  float inv  = rsqrtf(var + 1e-5f);
  float* hr = h + (size_t)row * DM;
  hr[t]       = (a0 - mean) * inv * w[t]       + b[t];
  hr[t + 256] = (a1 - mean) * inv * w[t + 256] + b[t + 256];
}

// ---------------- WMMA f32 16x16x4: one wave computes a 16 x (16*NSUB) tile ----------------
// a_row : &A[m0 + (lane&15)][2*(lane>>4)]          (row-major, stride lda)
// w_row0: &W[n0 + (lane&15)][2*(lane>>4)]          (weights row-major N x K)
// w rows for sub-tile s are offset by s*16*ldw.
template<int NSUB>
__device__ __forceinline__ void wmma_16xN(const float* __restrict__ a_row,
                                          const float* __restrict__ w_row0,
                                          int ldw16, int K, v8f* acc) {
#pragma unroll 4
  for (int k = 0; k < K; k += 4) {
    v2f a = *(const v2f*)(a_row + k);
#pragma unroll
    for (int s = 0; s < NSUB; ++s) {
      v2f b = *(const v2f*)(w_row0 + (size_t)s * ldw16 + k);
      acc[s] = __builtin_amdgcn_wmma_f32_16x16x4_f32(false, a, false, b,
                                                     (short)0, acc[s], false, false);
    }
  }
}

// ---------------- in_proj: xz(16384x1024) = h(16384x512) @ W^T ----------------
__global__ void gemm_in_kernel(const float* __restrict__ Hm,
                               const float* __restrict__ Wt,
                               float* __restrict__ Out) {
  int lane = threadIdx.x & 31, wave = threadIdx.x >> 5;   // 8 waves stacked in M
  int m0 = blockIdx.x * 128 + wave * 16;
  int n0 = blockIdx.y * 64;
  int half = lane >> 4, idx = lane & 15;
  const float* a_row = Hm + (size_t)(m0 + idx) * EE + 2*half;
  const float* w_row = Wt + (size_t)(n0 + idx) * EE + 2*half;
  v8f acc[4] = {};
  wmma_16xN<4>(a_row, w_row, 16*EE, EE, acc);
#pragma unroll
  for (int s = 0; s < 4; ++s) {
    float* c = Out + (size_t)(m0 + 8*half) * LDXZ + n0 + s*16 + idx;
#pragma unroll
    for (int v = 0; v < 8; ++v) c[(size_t)v * LDXZ] = acc[s][v];
  }
}

// ---------------- depthwise causal conv (k=2) + SiLU, both directions ----------------
__global__ void conv_kernel(const float* __restrict__ xz,
                            const float* __restrict__ cw0, const float* __restrict__ cb0,
                            const float* __restrict__ cw1, const float* __restrict__ cb1,
                            float* __restrict__ xc0, float* __restrict__ xc1) {
  int dir = blockIdx.y;
  size_t idx = (size_t)blockIdx.x * blockDim.x + threadIdx.x;   // over ROWS*EE
  int d = (int)(idx & (EE - 1));
  size_t row = idx >> 9;                                        // b*L + t
  int t = (int)(row & (LL - 1));
  const float* cw = dir ? cw1 : cw0;
  const float* cb = dir ? cb1 : cb0;
  float cur = xz[row * LDXZ + d];
  float prev = 0.f;
  if (dir == 0) { if (t > 0)      prev = xz[(row - 1) * LDXZ + d]; }
  else          { if (t < LL - 1) prev = xz[(row + 1) * LDXZ + d]; }
  float v = cw[d*2 + 0] * prev + cw[d*2 + 1] * cur + cb[d];
  (dir ? xc1 : xc0)[idx] = silu_f(v);
}

// ---------------- x_proj: dbc(16384x48) = xc @ W^T (48x512) ----------------
__global__ void gemm_xproj_kernel(const float* __restrict__ Xc,
                                  const float* __restrict__ Wt,
                                  float* __restrict__ Out) {
  int lane = threadIdx.x & 31, wave = threadIdx.x >> 5;   // 8 waves stacked in M
  int m0 = blockIdx.x * 128 + wave * 16;
  int half = lane >> 4, idx = lane & 15;
  const float* a_row = Xc + (size_t)(m0 + idx) * EE + 2*half;
  const float* w_row = Wt + (size_t)idx * EE + 2*half;    // n0 = 0, N = 48
  v8f acc[3] = {};
  wmma_16xN<3>(a_row, w_row, 16*EE, EE, acc);
#pragma unroll
  for (int s = 0; s < 3; ++s) {
    float* c = Out + (size_t)(m0 + 8*half) * LDBC + s*16 + idx;
#pragma unroll
    for (int v = 0; v < 8; ++v) c[(size_t)v * LDBC] = acc[s][v];
  }
}

// ---------------- dt_proj: delta = softplus(dt @ W^T + bias) ----------------
__global__ void gemm_dt_kernel(const float* __restrict__ Dbc,
                               const float* __restrict__ Wt,
                               const float* __restrict__ bias,
                               float* __restrict__ Delta) {
  int lane = threadIdx.x & 31, wave = threadIdx.x >> 5;
  int m0 = blockIdx.x * 128 + wave * 16;
  int n0 = blockIdx.y * 64;
  int half = lane >> 4, idx = lane & 15;
  const float* a_row = Dbc + (size_t)(m0 + idx) * LDBC + 2*half;  // dt = cols 0..31
  const float* w_row = Wt  + (size_t)(n0 + idx) * RR   + 2*half;
  v8f acc[4] = {};
  wmma_16xN<4>(a_row, w_row, 16*RR, RR, acc);
#pragma unroll
  for (int s = 0; s < 4; ++s) {
    float bv = bias[n0 + s*16 + idx];
    float* c = Delta + (size_t)(m0 + 8*half) * EE + n0 + s*16 + idx;
#pragma unroll
    for (int v = 0; v < 8; ++v) {
      float val = acc[s][v] + bv;
      c[(size_t)v * EE] = (val > 20.f) ? val : log1pf(__expf(val));
    }
  }
}

// ---------------- chunked selective scan ----------------
// Pass A: per (b,dir,d,chunk) compute sum(dt) and local final state S (h starting at 0).
__global__ void scan_chunk_kernel(const float* __restrict__ delta, const float* __restrict__ u,
                                  const float* __restrict__ dbc,
                                  const float* __restrict__ A_log,
                                  float* __restrict__ SDT, float* __restrict__ Sbuf, int dir) {
  int lane = threadIdx.x;
  int d = blockIdx.x * 32 + lane;
  int b = blockIdx.y;
  int c = blockIdx.z;
  float A[NS], h[NS];
#pragma unroll
  for (int n = 0; n < NS; ++n) { A[n] = -__expf(A_log[d*NS + n]); h[n] = 0.f; }
  float sdt = 0.f;
  size_t base = (size_t)b * LL;
  int j0 = c * CHUNK;
  for (int i = 0; i < CHUNK; ++i) {
    int j = j0 + i;
    int t = dir ? (LL - 1 - j) : j;
    size_t row = base + (size_t)t;
    if (i + 4 < CHUNK) {
      int jp = j + 4;
      int tp = dir ? (LL - 1 - jp) : jp;
      size_t rp = base + (size_t)tp;
      __builtin_prefetch(delta + rp*EE + d, 0, 1);
      __builtin_prefetch(u     + rp*EE + d, 0, 1);
      __builtin_prefetch(dbc   + rp*LDBC + RR, 0, 1);
    }
    float dt = delta[row*EE + d];
    float uu = u[row*EE + d];
    const float* bc = dbc + row*LDBC + RR;   // B: bc[0..7]
    float dtu = dt * uu;
    sdt += dt;
#pragma unroll
    for (int n = 0; n < NS; ++n) h[n] = __expf(dt * A[n]) * h[n] + dtu * bc[n];
  }
  int e = dir * BB + b;
  size_t sidx = (size_t)e * NCH + c;
  SDT[sidx * EE + d] = sdt;
#pragma unroll
  for (int n = 0; n < NS; ++n) Sbuf[(sidx * NS + n) * EE + d] = h[n];
}

// Pass B: sequentially compose chunk summaries -> initial state H0 for every chunk.
__global__ void scan_state_kernel(const float* __restrict__ A_log,
                                  const float* __restrict__ SDT,
                                  const float* __restrict__ Sbuf,
                                  float* __restrict__ H0, int dir) {
  int lane = threadIdx.x;
  int d = blockIdx.x * 32 + lane;
  int b = blockIdx.y;
  int e = dir * BB + b;
  float A[NS], h[NS];
#pragma unroll
  for (int n = 0; n < NS; ++n) { A[n] = -__expf(A_log[d*NS + n]); h[n] = 0.f; }
  for (int c = 0; c < NCH; ++c) {
    size_t sidx = (size_t)e * NCH + c;
#pragma unroll
    for (int n = 0; n < NS; ++n) H0[(sidx * NS + n) * EE + d] = h[n];
    float sdt = SDT[sidx * EE + d];
    float p = __expf(sdt);  // exp(sdt)^A? no: per-n below
    (void)p;
#pragma unroll
    for (int n = 0; n < NS; ++n)
      h[n] = __expf(A[n] * sdt) * h[n] + Sbuf[(sidx * NS + n) * EE + d];
  }
}

// Pass C: re-run each chunk from its correct initial state, emit gated outputs.
__global__ void scan_out_kernel(const float* __restrict__ delta, const float* __restrict__ u,
                                const float* __restrict__ dbc,   const float* __restrict__ xz,
                                const float* __restrict__ A_log, const float* __restrict__ Dp,
                                const float* __restrict__ H0,
                                float* __restrict__ yout, int dir) {
  int lane = threadIdx.x;
  int d = blockIdx.x * 32 + lane;
  int b = blockIdx.y;
  int c = blockIdx.z;
  int e = dir * BB + b;
  size_t sidx = (size_t)e * NCH + c;
  float A[NS], h[NS];
#pragma unroll
  for (int n = 0; n < NS; ++n) {
    A[n] = -__expf(A_log[d*NS + n]);
    h[n] = H0[(sidx * NS + n) * EE + d];
  }
  float Dd = Dp[d];
  size_t base = (size_t)b * LL;
  int j0 = c * CHUNK;
  for (int i = 0; i < CHUNK; ++i) {
    int j = j0 + i;
    int t = dir ? (LL - 1 - j) : j;
    size_t row = base + (size_t)t;
    if (i + 4 < CHUNK) {
      int jp = j + 4;
      int tp = dir ? (LL - 1 - jp) : jp;
      size_t rp = base + (size_t)tp;
      __builtin_prefetch(delta + rp*EE + d, 0, 1);
      __builtin_prefetch(u     + rp*EE + d, 0, 1);
      __builtin_prefetch(dbc   + rp*LDBC + RR, 0, 1);
      __builtin_prefetch(xz    + rp*LDXZ + EE + d, 0, 1);
    }
    float dt = delta[row*EE + d];
    float uu = u[row*EE + d];
    const float* bc = dbc + row*LDBC + RR;   // B: bc[0..7], C: bc[8..15]
    float dtu = dt * uu;
    float y = 0.f;
#pragma unroll
    for (int n = 0; n < NS; ++n) {
      h[n] = __expf(dt * A[n]) * h[n] + dtu * bc[n];
      y += h[n] * bc[NS + n];
    }
    y += uu * Dd;
    float zz = xz[row*LDXZ + EE + d];
    yout[row*EE + d] = 0.5f * y * silu_f(zz);
  }
}

// ---------------- combine y_f + y_b, RMSNorm * rms_w ----------------
__global__ void comb_kernel(const float* __restrict__ yf, const float* __restrict__ yb,
                            const float* __restrict__ rw, float* __restrict__ yn) {
  int row = blockIdx.x;
  int t = threadIdx.x;
  const float* f = yf + (size_t)row * EE;
  const float* g = yb + (size_t)row * EE;
  float a0 = f[t] + g[t];
  float a1 = f[t + 256] + g[t + 256];
  float ss = a0*a0 + a1*a1;
  __shared__ float lss[8];
  ss = waveReduceSum(ss);
  int wid = t >> 5, lane = t & 31;
  if (lane == 0) lss[wid] = ss;
  __syncthreads();
  if (wid == 0) {
    float s2 = (lane < 8) ? lss[lane] : 0.f;
    s2 = waveReduceSum(s2);
    if (lane == 0) lss[0] = s2;
  }
  __syncthreads();
  float inv = rsqrtf(lss[0] * (1.f/EE) + 1e-5f);
  float* o = yn + (size_t)row * EE;
  o[t]       = a0 * inv * rw[t];
  o[t + 256] = a1 * inv * rw[t + 256];
}

// ---------------- out_proj + residual ----------------
__global__ void gemm_out_kernel(const float* __restrict__ Yn,
                                const float* __restrict__ Wt,
                                const float* __restrict__ Xres,
                                float* __restrict__ Out) {
  int lane = threadIdx.x & 31, wave = threadIdx.x >> 5;
  int m0 = blockIdx.x * 128 + wave * 16;
  int n0 = blockIdx.y * 64;
  int half = lane >> 4, idx = lane & 15;
  const float* a_row = Yn + (size_t)(m0 + idx) * EE + 2*half;
  const float* w_row = Wt + (size_t)(n0 + idx) * DM + 2*half;
  v8f acc[4] = {};
  wmma_16xN<4>(a_row, w_row, 16*DM, EE, acc);
#pragma unroll
  for (int s = 0; s < 4; ++s) {
    size_t coff = (size_t)(m0 + 8*half) * DM + n0 + s*16 + idx;
    float* c = Out + coff;
    const float* xr = Xres + coff;
#pragma unroll
    for (int v = 0; v < 8; ++v) c[(size_t)v * DM] = acc[s][v] + xr[(size_t)v * DM];
  }
}

extern "C" void kernel_launch(void* const* d_in, const int* in_sizes, int n_in,
                              void* d_out, int out_size, void* d_ws, size_t ws_size,
                              hipStream_t stream) {
  (void)in_sizes; (void)n_in; (void)out_size; (void)ws_size;
  const float* x         = (const float*)d_in[0];
  const float* ln_w      = (const float*)d_in[1];
  const float* ln_b      = (const float*)d_in[2];
  const float* in_proj_w = (const float*)d_in[3];
  const float* conv_w    = (const float*)d_in[4];
  const float* conv_b    = (const float*)d_in[5];
  const float* x_proj_w  = (const float*)d_in[6];
  const float* dt_proj_w = (const float*)d_in[7];
  const float* dt_proj_b = (const float*)d_in[8];
  const float* A_log     = (const float*)d_in[9];
  const float* Dvec      = (const float*)d_in[10];
  const float* conv_w_b  = (const float*)d_in[11];
  const float* conv_b_b  = (const float*)d_in[12];
  const float* x_proj_wb = (const float*)d_in[13];
  const float* dt_proj_wb= (const float*)d_in[14];
  const float* dt_proj_bb= (const float*)d_in[15];
  const float* A_log_b   = (const float*)d_in[16];
  const float* D_b       = (const float*)d_in[17];
  const float* rms_w     = (const float*)d_in[18];
  const float* out_proj_w= (const float*)d_in[19];
  float* out = (float*)d_out;

  float* ws = (float*)d_ws;
  size_t o = 0;
  float* h    = ws + o; o += (size_t)ROWS * EE;      // 32MB (reused as yn)
  float* xz   = ws + o; o += (size_t)ROWS * LDXZ;    // 64MB
  float* xc0  = ws + o; o += (size_t)ROWS * EE;
  float* xc1  = ws + o; o += (size_t)ROWS * EE;
  float* dbc0 = ws + o; o += (size_t)ROWS * LDBC;
  float* dbc1 = ws + o; o += (size_t)ROWS * LDBC;
  float* del0 = ws + o; o += (size_t)ROWS * EE;
  float* del1 = ws + o; o += (size_t)ROWS * EE;
  float* y0   = ws + o; o += (size_t)ROWS * EE;
  float* y1   = ws + o; o += (size_t)ROWS * EE;
  float* sdt  = ws + o; o += (size_t)2 * BB * NCH * EE;           // 1MB
  float* sbuf = ws + o; o += (size_t)2 * BB * NCH * NS * EE;      // 8MB
  float* h0b  = ws + o; o += (size_t)2 * BB * NCH * NS * EE;      // 8MB

  ln_kernel<<<ROWS, 256, 0, stream>>>(x, ln_w, ln_b, h);
  gemm_in_kernel<<<dim3(ROWS/128, (2*EE)/64), 256, 0, stream>>>(h, in_proj_w, xz);
  conv_kernel<<<dim3((size_t)ROWS*EE/256, 2), 256, 0, stream>>>(
      xz, conv_w, conv_b, conv_w_b, conv_b_b, xc0, xc1);
  gemm_xproj_kernel<<<dim3(ROWS/128), 256, 0, stream>>>(xc0, x_proj_w,  dbc0);
  gemm_xproj_kernel<<<dim3(ROWS/128), 256, 0, stream>>>(xc1, x_proj_wb, dbc1);
  gemm_dt_kernel<<<dim3(ROWS/128, EE/64), 256, 0, stream>>>(dbc0, dt_proj_w,  dt_proj_b,  del0);
  gemm_dt_kernel<<<dim3(ROWS/128, EE/64), 256, 0, stream>>>(dbc1, dt_proj_wb, dt_proj_bb, del1);
  // chunked scan: pass A (chunk summaries), pass B (compose), pass C (emit)
  scan_chunk_kernel<<<dim3(EE/32, BB, NCH), 32, 0, stream>>>(del0, xc0, dbc0, A_log,   sdt, sbuf, 0);
  scan_chunk_kernel<<<dim3(EE/32, BB, NCH), 32, 0, stream>>>(del1, xc1, dbc1, A_log_b, sdt, sbuf, 1);
  scan_state_kernel<<<dim3(EE/32, BB), 32, 0, stream>>>(A_log,   sdt, sbuf, h0b, 0);
  scan_state_kernel<<<dim3(EE/32, BB), 32, 0, stream>>>(A_log_b, sdt, sbuf, h0b, 1);
  scan_out_kernel<<<dim3(EE/32, BB, NCH), 32, 0, stream>>>(del0, xc0, dbc0, xz, A_log,   Dvec, h0b, y0, 0);
  scan_out_kernel<<<dim3(EE/32, BB, NCH), 32, 0, stream>>>(del1, xc1, dbc1, xz, A_log_b, D_b,  h0b, y1, 1);
  comb_kernel<<<ROWS, 256, 0, stream>>>(y0, y1, rms_w, h);
  gemm_out_kernel<<<dim3(ROWS/128, DM/64), 256, 0, stream>>>(h, out_proj_w, x, out);
}